// APLoss_4020089389405
// MI455X (gfx1250) — compile-verified
//
#include <hip/hip_runtime.h>

typedef __attribute__((ext_vector_type(2))) float v2f;
typedef __attribute__((ext_vector_type(8))) float v8f;

#define B_   4
#define C_   128
#define H_   48
#define W_   48
#define HW_  2304            // 48*48
#define NQ_  20
#define STRIPS_ 144          // HW_/16
#define A_SLOPE 9.5f         // (NQ-1)/(MAXV-MINV)

// ---------------------------------------------------------------------------
// Kernel 1: bilinear grid_sample of d2 -> dbT2, transposed + K-pair packed:
//   dbT2[b][k2][j] = { db[j][2*k2], db[j][2*k2+1] }   (v2f, j fastest)
// One thread per (b, k2, j); total = 4 * 64 * 2304 = 589824 threads exactly.
// ---------------------------------------------------------------------------
__global__ __launch_bounds__(256) void aploss_grid_sample(
    const float* __restrict__ d2, const float* __restrict__ grid,
    v2f* __restrict__ dbT2) {
  int t = blockIdx.x * 256 + threadIdx.x;
  const int perImg = (C_ / 2) * HW_;
  int b   = t / perImg;
  int rem = t - b * perImg;
  int k2  = rem / HW_;
  int j   = rem - k2 * HW_;

  const float* g = grid + ((size_t)b * HW_ + j) * 2;
  float gx = (g[0] + 1.0f) * (W_ * 0.5f) - 0.5f;
  float gy = (g[1] + 1.0f) * (H_ * 0.5f) - 0.5f;
  float x0f = floorf(gx), y0f = floorf(gy);
  float wx1 = gx - x0f, wx0 = 1.0f - wx1;
  float wy1 = gy - y0f, wy0 = 1.0f - wy1;
  int x0 = (int)x0f, y0 = (int)y0f;
  int x1 = x0 + 1,   y1 = y0 + 1;
  float w00 = wx0 * wy0, w10 = wx1 * wy0, w01 = wx0 * wy1, w11 = wx1 * wy1;

  const float* p0 = d2 + (size_t)b * C_ * HW_ + (size_t)(2 * k2) * HW_;
  const float* p1 = p0 + HW_;

  auto fetch = [](const float* p, int xx, int yy) -> float {
    if (xx < 0 || xx >= W_ || yy < 0 || yy >= H_) return 0.0f;
    return p[yy * W_ + xx];
  };

  float a0 = fetch(p0, x0, y0) * w00 + fetch(p0, x1, y0) * w10 +
             fetch(p0, x0, y1) * w01 + fetch(p0, x1, y1) * w11;
  float a1 = fetch(p1, x0, y0) * w00 + fetch(p1, x1, y0) * w10 +
             fetch(p1, x0, y1) * w01 + fetch(p1, x1, y1) * w11;

  v2f o; o.x = a0; o.y = a1;
  dbT2[(size_t)b * perImg + (size_t)k2 * HW_ + j] = o;
}

// ---------------------------------------------------------------------------
// Kernel 2: fused scores-GEMM (V_WMMA_F32_16X16X4_F32) + 2-bin triangular
// scatter + per-row quantized-AP finalize. One block per (image, 16-row strip).
// 8 waves each own a 16-col j-tile per iteration and a private LDS histogram
// slab (16 rows x 20 bins x {nbs,rec}); slabs are reduced at the end.
// ---------------------------------------------------------------------------
__global__ __launch_bounds__(256) void aploss_strip(
    const float* __restrict__ d1, const int* __restrict__ mask,
    const float* __restrict__ rel, const v2f* __restrict__ dbT2,
    float* __restrict__ partial) {
  __shared__ v2f   As2[64 * 16];          // A tile: K-pair major, 16 rows
  __shared__ float accLds[8 * 16 * 40];   // per-wave [row][bin 0..19=nbs, 20..39=rec]
  __shared__ float finalAcc[16 * 40];
  __shared__ float lossBuf[16];

  const int tid   = threadIdx.x;
  const int blk   = blockIdx.x;
  const int b     = blk / STRIPS_;
  const int strip = blk - b * STRIPS_;
  const int i0    = strip * 16;

  // Load the 16x128 query tile (d1 is (C,HW): row i, chan c at c*HW + i).
  const float* d1b = d1 + (size_t)b * C_ * HW_;
  for (int e = tid; e < 64 * 16; e += 256) {
    int k2 = e >> 4, m = e & 15;
    v2f a;
    a.x = d1b[(size_t)(2 * k2)     * HW_ + i0 + m];
    a.y = d1b[(size_t)(2 * k2 + 1) * HW_ + i0 + m];
    As2[e] = a;
  }
  for (int e = tid; e < 8 * 640; e += 256) accLds[e] = 0.0f;
  __syncthreads();

  const int wave   = tid >> 5;
  const int lane   = tid & 31;
  const int m      = lane & 15;            // M for A-frag, N for B-frag
  const int hk     = (lane < 16) ? 0 : 1;  // K half: lanes 0-15 K0/K1, 16-31 K2/K3
  const int rowOff = (lane < 16) ? 0 : 8;  // C/D layout: vgpr r holds rows r, r+8

  const v2f* dbImg = dbT2 + (size_t)b * (size_t)(64 * HW_);
  const int* maskB = mask + b * HW_;
  float*     accW  = accLds + wave * 640;

  for (int jt = wave; jt < STRIPS_; jt += 8) {
    const int j0 = jt * 16;
    const int j  = j0 + m;

    v8f acc = {};
#pragma unroll
    for (int kk = 0; kk < 32; ++kk) {            // K = 128 = 32 * 4
      v2f a  = As2[(kk * 2 + hk) * 16 + m];
      v2f bb = dbImg[(size_t)(kk * 2 + hk) * HW_ + j];
      acc = __builtin_amdgcn_wmma_f32_16x16x4_f32(
          false, a, false, bb, (short)0, acc, false, false);
    }

    // Epilogue: triangular kernel hits exactly bins {floor(t), floor(t)+1}.
    const bool mw = (maskB[j] == 1);
    if (mw) {
      const int xj = j / W_;
      const int yj = j - xj * W_;
#pragma unroll
      for (int r = 0; r < 8; ++r) {
        const int irow = i0 + rowOff + r;
        const float s  = acc[r];
        const int xi = irow / W_;
        const int yi = irow - xi * W_;
        int dx = xi - xj; if (dx < 0) dx = -dx;
        int dy = yi - yj; if (dy < 0) dy = -dy;
        const bool lab = (dx <= 4) && (dy <= 4);

        float t = (1.0f - s) * A_SLOPE;          // t = A*(MAXV - s)
        t = fminf(fmaxf(t, 0.0f), 19.0f);
        int k0 = (int)t; if (k0 > 18) k0 = 18;
        const float w1 = t - (float)k0;
        const float w0 = 1.0f - w1;

        float* base = accW + (rowOff + r) * 40 + k0;
        atomicAdd(base,     w0);                 // nbs
        atomicAdd(base + 1, w1);
        if (lab) {
          atomicAdd(base + 20, w0);              // rec
          atomicAdd(base + 21, w1);
        }
      }
    }
  }
  __syncthreads();

  // Reduce the 8 per-wave slabs.
  for (int e = tid; e < 640; e += 256) {
    float s = 0.0f;
#pragma unroll
    for (int w = 0; w < 8; ++w) s += accLds[w * 640 + e];
    finalAcc[e] = s;
  }
  __syncthreads();

  // Per-row quantized AP + reliability-weighted loss.
  if (tid < 16) {
    const float* fa = finalAcc + tid * 40;
    float cn = 0.0f, cr = 0.0f, apn = 0.0f;
#pragma unroll
    for (int k = 0; k < NQ_; ++k) {
      const float nk = fa[k];
      const float rk = fa[20 + k];
      cn += nk; cr += rk;
      apn += rk * (cr / (1e-16f + cn));          // prec * rec
    }
    const float ap = apn / (1e-16f + cr);        // / sum(rec)
    const float rv = rel[b * HW_ + i0 + tid];
    lossBuf[tid] = 1.0f - (ap * rv + 0.5f * (1.0f - rv));
  }
  __syncthreads();
  if (tid == 0) {
    float s = 0.0f;
#pragma unroll
    for (int q = 0; q < 16; ++q) s += lossBuf[q];
    partial[blk] = s;
  }
}

// ---------------------------------------------------------------------------
// Kernel 3: deterministic final reduction of 576 strip partials -> mean.
// ---------------------------------------------------------------------------
__global__ __launch_bounds__(256) void aploss_reduce(
    const float* __restrict__ partial, float* __restrict__ out) {
  __shared__ float buf[256];
  const int tid = threadIdx.x;
  float s = 0.0f;
  for (int i = tid; i < B_ * STRIPS_; i += 256) s += partial[i];
  buf[tid] = s;
  __syncthreads();
  for (int off = 128; off > 0; off >>= 1) {
    if (tid < off) buf[tid] += buf[tid + off];
    __syncthreads();
  }
  if (tid == 0) out[0] = buf[0] * (1.0f / (float)(B_ * HW_));
}

// ---------------------------------------------------------------------------
extern "C" void kernel_launch(void* const* d_in, const int* in_sizes, int n_in,
                              void* d_out, int out_size, void* d_ws, size_t ws_size,
                              hipStream_t stream) {
  const float* d1   = (const float*)d_in[0];   // (4,128,48,48)
  const float* d2   = (const float*)d_in[1];   // (4,128,48,48)
  const float* rel  = (const float*)d_in[2];   // (4,1,48,48)
  const float* grid = (const float*)d_in[3];   // (4,48,48,2)
  const int*   mask = (const int*)d_in[4];     // (4,48,48)
  float* out = (float*)d_out;

  v2f* dbT2 = (v2f*)d_ws;                                       // 4*64*2304 v2f = 4.72 MB
  float* partial =
      (float*)((char*)d_ws + (size_t)B_ * 64 * HW_ * sizeof(v2f)); // 576 floats

  // 1) grid_sample: 4*64*2304 threads = 2304 blocks of 256 (exact).
  aploss_grid_sample<<<(B_ * (C_ / 2) * HW_) / 256, 256, 0, stream>>>(d2, grid, dbT2);
  // 2) fused WMMA GEMM + AP: one block per (image, 16-row strip).
  aploss_strip<<<B_ * STRIPS_, 256, 0, stream>>>(d1, mask, rel, dbT2, partial);
  // 3) final deterministic mean.
  aploss_reduce<<<1, 256, 0, stream>>>(partial, out);
}